// UserEmbeddingGenerator_3521873183303
// MI455X (gfx1250) — compile-verified
//
#include <hip/hip_runtime.h>

typedef float v2f __attribute__((ext_vector_type(2)));
typedef float v8f __attribute__((ext_vector_type(8)));

#define N_NODES 50000
#define FEAT    128   // feature width entering each scatter (d_in = d_hid = 128)

// ---------------------------------------------------------------------------
// zero a contiguous float range (grid-stride)
// ---------------------------------------------------------------------------
__global__ void sage_zero_kernel(float* __restrict__ p, long long n) {
  long long i = (long long)blockIdx.x * blockDim.x + threadIdx.x;
  long long stride = (long long)gridDim.x * blockDim.x;
  for (; i < n; i += stride) p[i] = 0.0f;
}

// ---------------------------------------------------------------------------
// edge-parallel scatter-add: msg[dst] += x[src]  (128 floats per edge)
// one thread handles one (edge, float4-chunk); 32 chunks of 4 floats = 128
// deg[dst] += 1 done by chunk-0 thread when addDeg != 0
// ---------------------------------------------------------------------------
__global__ void sage_scatter_kernel(const float* __restrict__ x,
                                    const int* __restrict__ src,
                                    const int* __restrict__ dst,
                                    float* __restrict__ msg,
                                    float* __restrict__ deg,
                                    int E, int addDeg) {
  long long gid = (long long)blockIdx.x * blockDim.x + threadIdx.x;
  long long total = (long long)E * 32;
  if (gid >= total) return;
  int e = (int)(gid >> 5);
  int c = (int)(gid & 31);
  int s = src[e];
  int d = dst[e];
  const float4* xs = (const float4*)(x + (long long)s * FEAT);
  float4 v = xs[c];
  float* p = msg + (long long)d * FEAT + c * 4;
  unsafeAtomicAdd(p + 0, v.x);
  unsafeAtomicAdd(p + 1, v.y);
  unsafeAtomicAdd(p + 2, v.z);
  unsafeAtomicAdd(p + 3, v.w);
  if (addDeg && c == 0) unsafeAtomicAdd(deg + d, 1.0f);
}

// ---------------------------------------------------------------------------
// fused SAGE layer GEMM on WMMA f32 16x16x4:
//   out[16x16 tile] = (msg/max(deg,1)) @ Wl + bl + xin @ Wr   (+ optional relu)
// one wave per 16x16 output tile. K, N are multiples of 16 (K mult of 4).
// A frag (16x4 f32): lane&15 = row, kb = (lane>>4)*2 -> {a.x: K=kb, a.y: K=kb+1}
// B frag (4x16 f32): lane&15 = col, same kb pairing
// C/D (v8f): VGPR i holds row i + 8*(lane>>4), col = lane&15
// ---------------------------------------------------------------------------
__global__ void sage_gemm_kernel(const float* __restrict__ msg,
                                 const float* __restrict__ deg,
                                 const float* __restrict__ xin,
                                 const float* __restrict__ Wl,
                                 const float* __restrict__ bl,
                                 const float* __restrict__ Wr,
                                 float* __restrict__ out,
                                 int K, int N, int doRelu) {
  const int lane = threadIdx.x & 31;
  const int wave = threadIdx.x >> 5;
  const int wavesPerBlock = blockDim.x >> 5;
  const int tile = blockIdx.x * wavesPerBlock + wave;
  const int colTiles = N >> 4;
  const int rowTiles = N_NODES / 16;           // 3125 (exact)
  const int rowTile = tile / colTiles;
  const int colTile = tile - rowTile * colTiles;
  if (rowTile >= rowTiles) return;             // wave-uniform -> EXEC all-1 below

  const int r   = (rowTile << 4) + (lane & 15);   // A-row this lane supplies
  const int col = (colTile << 4) + (lane & 15);   // B/C/D column this lane holds
  const int kb  = (lane >> 4) << 1;               // K-pair base (0 or 2)

  const float scale = 1.0f / fmaxf(deg[r], 1.0f); // mean aggregation divisor

  v8f c = {};
  const float* arow = msg + (long long)r * K;
  for (int k0 = 0; k0 < K; k0 += 4) {
    v2f a, b;
    a.x = arow[k0 + kb] * scale;
    a.y = arow[k0 + kb + 1] * scale;
    b.x = Wl[(k0 + kb) * N + col];
    b.y = Wl[(k0 + kb + 1) * N + col];
    c = __builtin_amdgcn_wmma_f32_16x16x4_f32(false, a, false, b,
                                              (short)0, c, false, false);
  }
  const float* xrow = xin + (long long)r * K;
  for (int k0 = 0; k0 < K; k0 += 4) {
    v2f a, b;
    a.x = xrow[k0 + kb];
    a.y = xrow[k0 + kb + 1];
    b.x = Wr[(k0 + kb) * N + col];
    b.y = Wr[(k0 + kb + 1) * N + col];
    c = __builtin_amdgcn_wmma_f32_16x16x4_f32(false, a, false, b,
                                              (short)0, c, false, false);
  }

  const float bias = bl[col];
  const int rbase = (rowTile << 4) + ((lane >> 4) << 3);
#pragma unroll
  for (int i = 0; i < 8; ++i) {
    float v = c[i] + bias;
    if (doRelu) v = fmaxf(v, 0.0f);
    out[(long long)(rbase + i) * N + col] = v;
  }
}

// ---------------------------------------------------------------------------
// host-side launch
// inputs: x, edge_index, W_l1, b_l1, W_r1, W_l2, b_l2, W_r2
// workspace layout: msg[N*128] | deg[N] | h1[N*128]
// ---------------------------------------------------------------------------
extern "C" void kernel_launch(void* const* d_in, const int* in_sizes, int n_in,
                              void* d_out, int out_size, void* d_ws, size_t ws_size,
                              hipStream_t stream) {
  const float* x   = (const float*)d_in[0];
  const int*   ei  = (const int*)d_in[1];
  const float* Wl1 = (const float*)d_in[2];
  const float* bl1 = (const float*)d_in[3];
  const float* Wr1 = (const float*)d_in[4];
  const float* Wl2 = (const float*)d_in[5];
  const float* bl2 = (const float*)d_in[6];
  const float* Wr2 = (const float*)d_in[7];
  float* out = (float*)d_out;

  const int E = in_sizes[1] / 2;
  const int* src = ei;
  const int* dst = ei + E;

  float* msg = (float*)d_ws;
  float* deg = msg + (long long)N_NODES * FEAT;
  float* h1  = deg + N_NODES;

  const int ZB = 2048, TB = 256;
  const long long scatterThreads = (long long)E * 32;
  const int scatterBlocks = (int)((scatterThreads + TB - 1) / TB);

  // ---- layer 1 ----
  sage_zero_kernel<<<ZB, TB, 0, stream>>>(msg, (long long)N_NODES * FEAT + N_NODES);
  sage_scatter_kernel<<<scatterBlocks, TB, 0, stream>>>(x, src, dst, msg, deg, E, 1);
  {
    const int tiles = (N_NODES / 16) * (128 / 16);   // 25000
    const int blocks = (tiles + 7) / 8;              // 8 waves / block
    sage_gemm_kernel<<<blocks, TB, 0, stream>>>(msg, deg, x, Wl1, bl1, Wr1, h1,
                                                128, 128, 1);
  }

  // ---- layer 2 ----
  sage_zero_kernel<<<ZB, TB, 0, stream>>>(msg, (long long)N_NODES * FEAT);
  sage_scatter_kernel<<<scatterBlocks, TB, 0, stream>>>(h1, src, dst, msg, deg, E, 0);
  {
    const int tiles = (N_NODES / 16) * (64 / 16);    // 12500
    const int blocks = (tiles + 7) / 8;
    sage_gemm_kernel<<<blocks, TB, 0, stream>>>(msg, deg, h1, Wl2, bl2, Wr2, out,
                                                128, 64, 0);
  }
}